// TemporalEncoding_57672820850797
// MI455X (gfx1250) — compile-verified
//
#include <hip/hip_runtime.h>

// LIF temporal encoding, MI455X (gfx1250, wave32).
// Store-bandwidth-bound: 80 MB out / 2 MB in -> optimize purely for the
// store stream. Each lane owns 4 consecutive floats (b128 traffic), keeps
// the membrane potential in 4 VGPRs, fully unrolls the 20 timesteps, and
// streams spikes + potentials with non-temporal (write-once) hints.

#define T_STEPS 20
#define BATCH 64
#define FEAT 8192
#define N_ELEMS (BATCH * FEAT)          // 524288 floats
#define DECAY 0.9f
#define THRESH 1.0f

typedef __attribute__((ext_vector_type(4))) float v4f;

__global__ __launch_bounds__(256) void
TemporalEncoding_lif_kernel(const float* __restrict__ x, float* __restrict__ out) {
    // float4 slot this thread owns
    const unsigned idx4 = blockIdx.x * blockDim.x + threadIdx.x;
    const unsigned i = idx4 * 4u;
    if (i >= N_ELEMS) return;

    // Input: one coalesced b128 load, lives in registers for all 20 steps.
    const v4f xv = *reinterpret_cast<const v4f*>(x + i);

    float* __restrict__ spike_base = out;                              // [T][N]
    float* __restrict__ mem_base   = out + (size_t)T_STEPS * N_ELEMS;  // [T][N]

    v4f mem = (v4f)0.0f;

#pragma unroll
    for (int t = 0; t < T_STEPS; ++t) {
        // mem = decay*mem + x  (v_fma per component)
        mem.x = __builtin_fmaf(DECAY, mem.x, xv.x);
        mem.y = __builtin_fmaf(DECAY, mem.y, xv.y);
        mem.z = __builtin_fmaf(DECAY, mem.z, xv.z);
        mem.w = __builtin_fmaf(DECAY, mem.w, xv.w);

        // spike = mem >= thr ? 1 : 0  (v_cmp + v_cndmask)
        v4f sp;
        sp.x = (mem.x >= THRESH) ? 1.0f : 0.0f;
        sp.y = (mem.y >= THRESH) ? 1.0f : 0.0f;
        sp.z = (mem.z >= THRESH) ? 1.0f : 0.0f;
        sp.w = (mem.w >= THRESH) ? 1.0f : 0.0f;

        const size_t off = (size_t)t * N_ELEMS + i;

        // Write-once output: non-temporal b128 streaming stores (gfx1250 TH hint),
        // two independent 40 MB streams, never re-read by the kernel.
        __builtin_nontemporal_store(sp,  reinterpret_cast<v4f*>(spike_base + off));
        __builtin_nontemporal_store(mem, reinterpret_cast<v4f*>(mem_base   + off));

        // reset: mem -= spike * thr  (thr == 1)
        mem.x -= sp.x;
        mem.y -= sp.y;
        mem.z -= sp.z;
        mem.w -= sp.w;
    }
}

extern "C" void kernel_launch(void* const* d_in, const int* in_sizes, int n_in,
                              void* d_out, int out_size, void* d_ws, size_t ws_size,
                              hipStream_t stream) {
    (void)in_sizes; (void)n_in; (void)out_size; (void)d_ws; (void)ws_size;
    const float* x = (const float*)d_in[0];
    float* out = (float*)d_out;

    // 524288 floats / 4 per thread = 131072 threads = 512 blocks x 256 (8 wave32/WG)
    const int threads = 256;
    const int blocks = (N_ELEMS / 4 + threads - 1) / threads;
    TemporalEncoding_lif_kernel<<<blocks, threads, 0, stream>>>(x, out);
}